// DilatedResBlock_64707977282333
// MI455X (gfx1250) — compile-verified
//
#include <hip/hip_runtime.h>

typedef __attribute__((ext_vector_type(2))) float v2f;
typedef __attribute__((ext_vector_type(8))) float v8f;

#define NPTS 32768          // N
#define NBATCH 2            // B
#define TOTPTS (NBATCH*NPTS)
#define LSTR 65             // LDS row stride for 16x64 tiles (avoids bank conflicts)

// ---------------------------------------------------------------------------
// Kernel A: y = relu(feats @ W_res + b_res) -> d_out   (65536 x 128)
//           x = relu(feats @ W0    + b0)    -> ws       (65536 x 32)
// One wave handles a 16-row slice; fp32 WMMA 16x16x4, K=16 -> 4 chained WMMAs
// per 16-col tile.
// ---------------------------------------------------------------------------
template<int NC>
__device__ __forceinline__ void gemm_tile16(const float* __restrict__ W,
                                            const float* __restrict__ bias,
                                            float* __restrict__ dst,
                                            int row0, int n0,
                                            const v2f a[4], int m, int hi) {
  v8f acc = {};
#pragma unroll
  for (int f = 0; f < 4; ++f) {
    int k = 4 * f + 2 * hi;                 // B 4x16 frag: VGPR0=K0/K2, VGPR1=K1/K3
    v2f bf;
    bf.x = W[k * NC + n0 + m];
    bf.y = W[(k + 1) * NC + n0 + m];
    acc = __builtin_amdgcn_wmma_f32_16x16x4_f32(
        false, a[f], false, bf, (short)0, acc, false, false);
  }
  float bv = bias[n0 + m];
#pragma unroll
  for (int r = 0; r < 8; ++r) {             // C/D: VGPR r -> M = r (+8 for hi half)
    float v = acc[r] + bv;
    v = v > 0.f ? v : 0.f;                  // relu
    dst[(size_t)(row0 + hi * 8 + r) * NC + n0 + m] = v;
  }
}

__global__ void __launch_bounds__(256)
in_gemm_kernel(const float* __restrict__ feats,
               const float* __restrict__ W0,   const float* __restrict__ b0,
               const float* __restrict__ Wres, const float* __restrict__ bres,
               float* __restrict__ xbuf, float* __restrict__ ybuf) {
  int lane = threadIdx.x & 31;
  int wave = threadIdx.x >> 5;
  int row0 = (blockIdx.x * 8 + wave) * 16;
  int m = lane & 15, hi = lane >> 4;

  // A 16x4 f32 frag: lanes 0-15 hold (M=lane,K0/K1), lanes 16-31 (M=lane-16,K2/K3)
  const float* ar = feats + (size_t)(row0 + m) * 16;
  v2f a[4];
#pragma unroll
  for (int f = 0; f < 4; ++f) {
    int k = 4 * f + 2 * hi;
    a[f].x = ar[k];
    a[f].y = ar[k + 1];
  }

  gemm_tile16<32>(W0, b0, xbuf, row0, 0,  a, m, hi);
  gemm_tile16<32>(W0, b0, xbuf, row0, 16, a, m, hi);
#pragma unroll
  for (int t = 0; t < 8; ++t)
    gemm_tile16<128>(Wres, bres, ybuf, row0, t * 16, a, m, hi);
}

// ---------------------------------------------------------------------------
// Kernel B: per point -- gather K=16 neighbors, rppe, h=relu(rppe@Wl1),
// lo1=[n_feats|h] (16x64), s=lo1@Ws1+bs1 via WMMA, channel softmax, K-pool,
// relu(att@Wf1+bf1), relu(.@W1+b1), out = leaky_relu(x1 + y).
// 1 wave per point, 4 waves per block. att_0 branch is dead -> skipped.
// ---------------------------------------------------------------------------
__global__ void __launch_bounds__(128)
locse_att_kernel(const float* __restrict__ pc,
                 const int*   __restrict__ n_idx,
                 const float* __restrict__ xbuf,
                 const float* __restrict__ Wl1, const float* __restrict__ bl1,
                 const float* __restrict__ Ws1, const float* __restrict__ bs1,
                 const float* __restrict__ Wf1, const float* __restrict__ bf1,
                 const float* __restrict__ W1,  const float* __restrict__ b1,
                 float* __restrict__ out) {
  __shared__ float sWs1[64 * 64];           // 16 KB, shared by block
  __shared__ float slo[4][16 * LSTR];       // lo1 per wave
  __shared__ float ssc[4][16 * LSTR];       // scores (pre-softmax) per wave
  __shared__ float srp[4][16 * 8];          // rppe per wave
  __shared__ float satt[4][64];
  __shared__ float sxat[4][64];
  __shared__ float srmax[4][16];
  __shared__ float srinv[4][16];
  __shared__ int   sidx[4][16];

  int tid  = threadIdx.x;
  int lane = tid & 31;
  int wave = tid >> 5;
  int p = blockIdx.x * 4 + wave;            // flat b*N+n index
  int b = p >> 15;                          // N = 32768
  int m = lane & 15, hi = lane >> 4;

  for (int i = tid; i < 64 * 64; i += 128) sWs1[i] = Ws1[i];

  // ---- stage A: indices + relative point position encoding (lanes 0-15) ----
  if (lane < 16) {
    int k = lane;
    int idx = n_idx[(size_t)p * 16 + k];
    sidx[wave][k] = idx;
    float cx = pc[(size_t)p * 2 + 0], cy = pc[(size_t)p * 2 + 1];
    size_t q = (size_t)(b * NPTS + idx);
    float nx = pc[q * 2 + 0], ny = pc[q * 2 + 1];
    float rx = cx - nx, ry = cy - ny;
    float nr = sqrtf(rx * rx + ry * ry);
    float* rp = &srp[wave][k * 8];
    rp[0] = cx; rp[1] = cy; rp[2] = nx; rp[3] = ny;
    rp[4] = rx; rp[5] = ry; rp[6] = nr;
  }
  __syncthreads();

  // ---- stage B: gather n_feats (cols 0-31) + h = relu(rppe@Wl1) (cols 32-63) ----
  {
    int k = m;
    int idx = sidx[wave][k];
    const float* xr = xbuf + (size_t)(b * NPTS + idx) * 32 + hi * 16;
    float* lor = &slo[wave][k * LSTR];
#pragma unroll
    for (int t = 0; t < 16; ++t) lor[hi * 16 + t] = xr[t];

    const float* rp = &srp[wave][k * 8];
    float r0 = rp[0], r1 = rp[1], r2 = rp[2], r3 = rp[3];
    float r4 = rp[4], r5 = rp[5], r6 = rp[6];
#pragma unroll
    for (int t = 0; t < 16; ++t) {
      int j = hi * 16 + t;                  // 0..31 into Wl1/bl1
      float acc = bl1[j];
      acc += r0 * Wl1[0 * 32 + j] + r1 * Wl1[1 * 32 + j] + r2 * Wl1[2 * 32 + j]
           + r3 * Wl1[3 * 32 + j] + r4 * Wl1[4 * 32 + j] + r5 * Wl1[5 * 32 + j]
           + r6 * Wl1[6 * 32 + j];
      lor[32 + j] = acc > 0.f ? acc : 0.f;
    }
  }
  __syncthreads();

  // ---- stage C: s = lo1 @ Ws1 + bs1  (16x64, K=64) via fp32 WMMA ----
  {
    v2f af[16];
    const float* lr = &slo[wave][m * LSTR];
#pragma unroll
    for (int f = 0; f < 16; ++f) {
      int k = 4 * f + 2 * hi;
      af[f].x = lr[k];
      af[f].y = lr[k + 1];
    }
#pragma unroll
    for (int t = 0; t < 4; ++t) {
      int n0 = t * 16;
      v8f acc = {};
#pragma unroll
      for (int f = 0; f < 16; ++f) {
        int k = 4 * f + 2 * hi;
        v2f bf;
        bf.x = sWs1[k * 64 + n0 + m];
        bf.y = sWs1[(k + 1) * 64 + n0 + m];
        acc = __builtin_amdgcn_wmma_f32_16x16x4_f32(
            false, af[f], false, bf, (short)0, acc, false, false);
      }
      float bv = bs1[n0 + m];
#pragma unroll
      for (int r = 0; r < 8; ++r)
        ssc[wave][(hi * 8 + r) * LSTR + n0 + m] = acc[r] + bv;
    }
  }
  __syncthreads();

  // ---- stage D: per-row (per-neighbor) softmax stats over 64 channels ----
  if (lane < 16) {
    const float* sr = &ssc[wave][lane * LSTR];
    float mx = -3.402823466e+38f;
    for (int c = 0; c < 64; ++c) mx = fmaxf(mx, sr[c]);
    float sm = 0.f;
    for (int c = 0; c < 64; ++c) sm += __expf(sr[c] - mx);
    srmax[wave][lane] = mx;
    srinv[wave][lane] = 1.f / sm;
  }
  __syncthreads();

  // ---- stage E: att[c] = sum_k lo1[k,c] * softmax(s)[k,c] ----
#pragma unroll
  for (int h = 0; h < 2; ++h) {
    int c = lane + 32 * h;
    float acc = 0.f;
#pragma unroll
    for (int k = 0; k < 16; ++k) {
      float e = __expf(ssc[wave][k * LSTR + c] - srmax[wave][k]) * srinv[wave][k];
      acc += slo[wave][k * LSTR + c] * e;
    }
    satt[wave][c] = acc;
  }
  __syncthreads();

  // ---- stage F: xatt = relu(att @ Wf1 + bf1)  (64) ----
#pragma unroll
  for (int h = 0; h < 2; ++h) {
    int c = lane + 32 * h;
    float acc = bf1[c];
    for (int i = 0; i < 64; ++i) acc += satt[wave][i] * Wf1[i * 64 + c];
    sxat[wave][c] = acc > 0.f ? acc : 0.f;
  }
  __syncthreads();

  // ---- stage G: x1 = relu(xatt @ W1 + b1); out = leaky_relu(x1 + y, 0.2) ----
#pragma unroll
  for (int t = 0; t < 4; ++t) {
    int j = lane + 32 * t;
    float acc = b1[j];
    for (int i = 0; i < 64; ++i) acc += sxat[wave][i] * W1[i * 128 + j];
    acc = acc > 0.f ? acc : 0.f;
    float v = acc + out[(size_t)p * 128 + j];   // y was written by kernel A
    out[(size_t)p * 128 + j] = v > 0.f ? v : 0.2f * v;
  }
}

// ---------------------------------------------------------------------------
extern "C" void kernel_launch(void* const* d_in, const int* in_sizes, int n_in,
                              void* d_out, int out_size, void* d_ws, size_t ws_size,
                              hipStream_t stream) {
  (void)in_sizes; (void)n_in; (void)out_size; (void)ws_size;
  const float* pc    = (const float*)d_in[0];
  const float* feats = (const float*)d_in[1];
  const int*   n_idx = (const int*)  d_in[2];
  const float* W_res = (const float*)d_in[3];
  const float* b_res = (const float*)d_in[4];
  const float* W0    = (const float*)d_in[5];
  const float* b0    = (const float*)d_in[6];
  // d_in[7..12] = Wl0,bl0,Ws0,bs0,Wf0,bf0 -> att_0 branch is dead code, skipped
  const float* Wl1   = (const float*)d_in[13];
  const float* bl1   = (const float*)d_in[14];
  const float* Ws1   = (const float*)d_in[15];
  const float* bs1   = (const float*)d_in[16];
  const float* Wf1   = (const float*)d_in[17];
  const float* bf1   = (const float*)d_in[18];
  const float* W1    = (const float*)d_in[19];
  const float* b1    = (const float*)d_in[20];

  float* out  = (float*)d_out;
  float* xbuf = (float*)d_ws;               // 65536 x 32 f32 = 8 MB scratch

  // y -> d_out, x -> ws
  in_gemm_kernel<<<TOTPTS / 128, 256, 0, stream>>>(feats, W0, b0, W_res, b_res,
                                                   xbuf, out);
  // fused locSE_1 + att_1 + MLP_1 + residual
  locse_att_kernel<<<TOTPTS / 4, 128, 0, stream>>>(pc, n_idx, xbuf,
                                                   Wl1, bl1, Ws1, bs1,
                                                   Wf1, bf1, W1, b1, out);
}